// GCNLipMoE_47665547051798
// MI455X (gfx1250) — compile-verified
//
#include <hip/hip_runtime.h>
#include <hip/hip_bf16.h>

#define NODES 100000
#define NEDGE 800000
#define INF   512
#define HF    256
#define OUTF  64
#define NEXP  4

typedef __attribute__((ext_vector_type(2))) float v2f;
typedef __attribute__((ext_vector_type(8))) float v8f;

__device__ __forceinline__ v8f wmma_f32x4(v2f a, v2f b, v8f c) {
  // D = A(16x4 f32) * B(4x16 f32) + C(16x16 f32)
  return __builtin_amdgcn_wmma_f32_16x16x4_f32(false, a, false, b, (short)0, c,
                                               false, false);
}

// ---------------- init: deg=1 (self loop), perm=-1, cnt=0 ----------------
__global__ void k_init(float* __restrict__ deg, int* __restrict__ perm,
                       int* __restrict__ cnt) {
  int i = blockIdx.x * 256 + threadIdx.x;
  if (i < NODES) deg[i] = 1.0f;
  if (i < NODES + 64) perm[i] = -1;
  if (i < NEXP) cnt[i] = 0;
}

// ---------------- in-degree over col ----------------
__global__ void k_deg(const int* __restrict__ ei, float* __restrict__ deg) {
  int e = blockIdx.x * 256 + threadIdx.x;
  if (e < NEDGE) atomicAdd(&deg[ei[NEDGE + e]], 1.0f);
}

__global__ void k_dinv(float* __restrict__ deg) {
  int i = blockIdx.x * 256 + threadIdx.x;
  if (i < NODES) deg[i] = rsqrtf(deg[i]);  // deg >= 1 always (self loop)
}

// ---------------- gating: argmax(x @ w_gate), slot assignment ----------------
__global__ __launch_bounds__(256) void k_gate(const float* __restrict__ x,
                                              const float* __restrict__ wg,
                                              int* __restrict__ eid,
                                              int* __restrict__ slot,
                                              int* __restrict__ cnt) {
  int n = blockIdx.x * 8 + (threadIdx.x >> 5);
  if (n >= NODES) return;
  int lane = threadIdx.x & 31;
  float a0 = 0.f, a1 = 0.f, a2 = 0.f, a3 = 0.f;
  const float* xr = x + (size_t)n * INF;
  for (int k = lane; k < INF; k += 32) {
    float xv = xr[k];
    const float* g = wg + k * 4;
    a0 += xv * g[0]; a1 += xv * g[1]; a2 += xv * g[2]; a3 += xv * g[3];
  }
  for (int m = 16; m >= 1; m >>= 1) {
    a0 += __shfl_xor(a0, m); a1 += __shfl_xor(a1, m);
    a2 += __shfl_xor(a2, m); a3 += __shfl_xor(a3, m);
  }
  if (lane == 0) {
    int best = 0; float bv = a0;
    if (a1 > bv) { bv = a1; best = 1; }
    if (a2 > bv) { bv = a2; best = 2; }
    if (a3 > bv) { bv = a3; best = 3; }
    eid[n] = best;
    slot[n] = atomicAdd(&cnt[best], 1);
  }
}

// 16-aligned per-expert segment offsets
__global__ void k_offsets(const int* __restrict__ cnt, int* __restrict__ offs) {
  if (threadIdx.x == 0 && blockIdx.x == 0) {
    int o = 0;
    offs[0] = 0;
    for (int e = 0; e < NEXP; ++e) {
      o += (cnt[e] + 15) & ~15;
      offs[e + 1] = o;
    }
  }
}

__global__ void k_scatter(const int* __restrict__ eid, const int* __restrict__ slot,
                          const int* __restrict__ offs, int* __restrict__ perm) {
  int n = blockIdx.x * 256 + threadIdx.x;
  if (n < NODES) perm[offs[eid[n]] + slot[n]] = n;
}

// ---------------- MoE GEMM: h1[perm tile] = x @ W_e + b_e (fp32 WMMA) ------
__global__ __launch_bounds__(256) void k_moe_gemm(const float* __restrict__ x,
                                                  const float* __restrict__ ew,
                                                  const float* __restrict__ eb,
                                                  const int* __restrict__ perm,
                                                  const int* __restrict__ offs,
                                                  float* __restrict__ h1) {
  __shared__ float ldsA[16][INF + 4];   // +4 pad: conflict-free column reads
  __shared__ int rowNode[16];
  const int start = blockIdx.x * 16;
  if (start >= offs[NEXP]) return;
  int e = 0;
  while (e < NEXP - 1 && start >= offs[e + 1]) ++e;
  const int tid = threadIdx.x;
  if (tid < 16) rowNode[tid] = perm[start + tid];
  __syncthreads();
  // stage 16x512 A tile (gathered rows; -1 rows -> zero)
  for (int i = tid; i < 16 * (INF / 4); i += 256) {
    int r = i / (INF / 4);
    int k4 = (i % (INF / 4)) * 4;
    int node = rowNode[r];
    float4 v = make_float4(0.f, 0.f, 0.f, 0.f);
    if (node >= 0) v = *(const float4*)(x + (size_t)node * INF + k4);
    ldsA[r][k4 + 0] = v.x; ldsA[r][k4 + 1] = v.y;
    ldsA[r][k4 + 2] = v.z; ldsA[r][k4 + 3] = v.w;
  }
  __syncthreads();
  const int wave = tid >> 5, lane = tid & 31;
  const int half = lane >> 4, ln = lane & 15;
  const int c0 = wave * 32;                       // 8 waves cover 256 cols
  const float* We = ew + (size_t)e * INF * HF;
  v8f acc0 = {}; v8f acc1 = {};
  for (int k0 = 0; k0 < INF; k0 += 4) {
    // A frag: lanes 0-15 -> K=k0,k0+1 ; lanes 16-31 -> K=k0+2,k0+3 (M=ln)
    v2f a;
    a.x = ldsA[ln][k0 + 2 * half];
    a.y = ldsA[ln][k0 + 2 * half + 1];
    // B frag: lane holds N=ln; vgpr0/1 = K and K+1 (halves offset by 2)
    const float* bp = We + (size_t)(k0 + 2 * half) * HF + c0 + ln;
    v2f b0, b1;
    b0.x = bp[0];  b0.y = bp[HF];
    b1.x = bp[16]; b1.y = bp[HF + 16];
    acc0 = wmma_f32x4(a, b0, acc0);
    acc1 = wmma_f32x4(a, b1, acc1);
  }
  // D layout: VGPR v -> M = v + 8*half, lane -> N = ln
  for (int vv = 0; vv < 8; ++vv) {
    int r = vv + 8 * half;
    int node = rowNode[r];
    if (node >= 0) {
      float* dst = h1 + (size_t)node * HF;
      dst[c0 + ln]      = acc0[vv] + eb[e * HF + c0 + ln];
      dst[c0 + 16 + ln] = acc1[vv] + eb[e * HF + c0 + 16 + ln];
    }
  }
}

// ---------------- zero fill (float4) ----------------
__global__ void k_zero(float* __restrict__ p, int n4) {
  int i = blockIdx.x * 256 + threadIdx.x;
  if (i < n4) ((float4*)p)[i] = make_float4(0.f, 0.f, 0.f, 0.f);
}

// ---------------- edge propagate: agg[col] += h[row]*dinv[row]*dinv[col] ----
template <int F>
__global__ __launch_bounds__(256) void k_prop(const int* __restrict__ ei,
                                              const float* __restrict__ dinv,
                                              const float* __restrict__ h,
                                              float* __restrict__ agg) {
  int e = blockIdx.x * 8 + (threadIdx.x >> 5);
  if (e >= NEDGE) return;
  int lane = threadIdx.x & 31;
  int r = ei[e], c = ei[NEDGE + e];
  float nrm = dinv[r] * dinv[c];
  const float* src = h + (size_t)r * F;
  float* dst = agg + (size_t)c * F;
#pragma unroll
  for (int f = lane; f < F; f += 32) atomicAdd(dst + f, src[f] * nrm);
}

// ---------------- self loop + bias1 + relu (in place on agg1) ----------------
__global__ void k_relu(const float* __restrict__ h1, const float* __restrict__ dinv,
                       const float* __restrict__ b1, float* __restrict__ agg1) {
  size_t i = (size_t)blockIdx.x * 256 + threadIdx.x;
  if (i >= (size_t)NODES * HF) return;
  int n = (int)(i / HF), c = (int)(i % HF);
  float dv = dinv[n];
  float v = agg1[i] + h1[i] * dv * dv + b1[c];
  agg1[i] = fmaxf(v, 0.0f);
}

// ---------------- GEMM2: h3 = relu_out @ w2 (fp32 WMMA, 16x64 per block) ----
__global__ __launch_bounds__(128) void k_gemm2(const float* __restrict__ hin,
                                               const float* __restrict__ w2,
                                               float* __restrict__ h3) {
  __shared__ float ldsA[16][HF + 4];
  const int row0 = blockIdx.x * 16;  // NODES % 16 == 0
  const int tid = threadIdx.x;
  for (int i = tid; i < 16 * (HF / 4); i += 128) {
    int r = i / (HF / 4);
    int k4 = (i % (HF / 4)) * 4;
    float4 v = *(const float4*)(hin + (size_t)(row0 + r) * HF + k4);
    ldsA[r][k4 + 0] = v.x; ldsA[r][k4 + 1] = v.y;
    ldsA[r][k4 + 2] = v.z; ldsA[r][k4 + 3] = v.w;
  }
  __syncthreads();
  const int wave = tid >> 5, lane = tid & 31;
  const int half = lane >> 4, ln = lane & 15;
  const int c0 = wave * 16;  // 4 waves cover 64 cols
  v8f acc = {};
  for (int k0 = 0; k0 < HF; k0 += 4) {
    v2f a;
    a.x = ldsA[ln][k0 + 2 * half];
    a.y = ldsA[ln][k0 + 2 * half + 1];
    const float* bp = w2 + (size_t)(k0 + 2 * half) * OUTF + c0 + ln;
    v2f b;
    b.x = bp[0]; b.y = bp[OUTF];
    acc = wmma_f32x4(a, b, acc);
  }
  for (int vv = 0; vv < 8; ++vv) {
    int r = vv + 8 * half;
    h3[(size_t)(row0 + r) * OUTF + c0 + ln] = acc[vv];
  }
}

// ---------------- self loop + bias2 + log_softmax over 64 ----------------
__global__ __launch_bounds__(256) void k_final(const float* __restrict__ h3,
                                               const float* __restrict__ agg2,
                                               const float* __restrict__ dinv,
                                               const float* __restrict__ b2,
                                               float* __restrict__ out) {
  int n = blockIdx.x * 8 + (threadIdx.x >> 5);
  if (n >= NODES) return;
  int lane = threadIdx.x & 31;
  float dv = dinv[n]; dv *= dv;
  size_t base = (size_t)n * OUTF;
  float v0 = agg2[base + lane]      + h3[base + lane]      * dv + b2[lane];
  float v1 = agg2[base + lane + 32] + h3[base + lane + 32] * dv + b2[lane + 32];
  float m = fmaxf(v0, v1);
  for (int s = 16; s >= 1; s >>= 1) m = fmaxf(m, __shfl_xor(m, s));
  float sum = expf(v0 - m) + expf(v1 - m);
  for (int s = 16; s >= 1; s >>= 1) sum += __shfl_xor(sum, s);
  float l = m + logf(sum);
  out[base + lane]      = v0 - l;
  out[base + lane + 32] = v1 - l;
}

extern "C" void kernel_launch(void* const* d_in, const int* in_sizes, int n_in,
                              void* d_out, int out_size, void* d_ws, size_t ws_size,
                              hipStream_t stream) {
  const float* x  = (const float*)d_in[0];
  const int*   ei = (const int*)d_in[1];
  const float* wg = (const float*)d_in[2];
  const float* ew = (const float*)d_in[3];
  const float* eb = (const float*)d_in[4];
  const float* b1 = (const float*)d_in[5];
  const float* w2 = (const float*)d_in[6];
  const float* b2 = (const float*)d_in[7];
  float* out = (float*)d_out;

  char* base = (char*)d_ws;
  size_t o = 0;
  auto take = [&](size_t bytes) {
    char* p = base + o;
    o = (o + bytes + 255) & ~(size_t)255;
    return p;
  };
  float* deg  = (float*)take((size_t)NODES * 4);            // becomes dinv
  int*   eid  = (int*)take((size_t)NODES * 4);
  int*   slot = (int*)take((size_t)NODES * 4);
  int*   cnt  = (int*)take(NEXP * 4);
  int*   offs = (int*)take((NEXP + 1) * 4);
  int*   perm = (int*)take((size_t)(NODES + 64) * 4);
  float* h1   = (float*)take((size_t)NODES * HF * 4);
  float* agg1 = (float*)take((size_t)NODES * HF * 4);
  float* h3   = (float*)take((size_t)NODES * OUTF * 4);
  float* agg2 = (float*)take((size_t)NODES * OUTF * 4);

  // normalization
  k_init<<<(NODES + 64 + 255) / 256, 256, 0, stream>>>(deg, perm, cnt);
  k_deg<<<(NEDGE + 255) / 256, 256, 0, stream>>>(ei, deg);
  k_dinv<<<(NODES + 255) / 256, 256, 0, stream>>>(deg);
  // MoE bucketing
  k_gate<<<NODES / 8, 256, 0, stream>>>(x, wg, eid, slot, cnt);
  k_offsets<<<1, 1, 0, stream>>>(cnt, offs);
  k_scatter<<<(NODES + 255) / 256, 256, 0, stream>>>(eid, slot, offs, perm);
  // conv1
  k_moe_gemm<<<(NODES + 64) / 16, 256, 0, stream>>>(x, ew, eb, perm, offs, h1);
  k_zero<<<(NODES * HF / 4 + 255) / 256, 256, 0, stream>>>(agg1, NODES * HF / 4);
  k_prop<HF><<<NEDGE / 8, 256, 0, stream>>>(ei, deg, h1, agg1);
  k_relu<<<NODES * HF / 256, 256, 0, stream>>>(h1, deg, b1, agg1);
  // conv2
  k_gemm2<<<NODES / 16, 256, 0, stream>>>(agg1, w2, h3);
  k_zero<<<(NODES * OUTF / 4 + 255) / 256, 256, 0, stream>>>(agg2, NODES * OUTF / 4);
  k_prop<OUTF><<<NEDGE / 8, 256, 0, stream>>>(ei, deg, h3, agg2);
  k_final<<<NODES / 8, 256, 0, stream>>>(h3, agg2, deg, b2, out);
}